// DeformConv_69587060130437
// MI455X (gfx1250) — compile-verified
//
#include <hip/hip_runtime.h>
#include <hip/hip_bf16.h>
#include <math.h>

typedef __attribute__((ext_vector_type(2))) float v2f;
typedef __attribute__((ext_vector_type(8))) float v8f;
typedef __attribute__((ext_vector_type(4))) unsigned int v4u;
typedef __attribute__((ext_vector_type(8))) int v8i;
typedef __attribute__((ext_vector_type(4))) int v4i;

#define B_  8
#define C_  128
#define S_  64
#define HW_ (S_ * S_)
#define NPIX_ (B_ * HW_)     // 32768
#define NOUT_ 18             // 2 * 3 * 3 offset channels
#define KTOT_ (C_ * 9)       // 1152
#define NSTEP_ 288           // 9 taps * 32 c-blocks of 4
#define BPACK_N (NSTEP_ * 128)   // 36864 floats = 147456 bytes

// TDM available? (this toolchain: 6-arg builtin). Fallback = plain copy.
#if defined(__has_builtin)
#  if __has_builtin(__builtin_amdgcn_tensor_load_to_lds)
#    define HAVE_TDM 1
#  else
#    define HAVE_TDM 0
#  endif
#else
#  define HAVE_TDM 0
#endif

// ---------------------------------------------------------------------------
// Generic batched tiled transpose: in [batch][R][C] -> out [batch][C][R]
// grid = (C/32, R/32, batch), block = (32, 8)
// ---------------------------------------------------------------------------
__global__ void transpose_kernel(const float* __restrict__ in,
                                 float* __restrict__ out, int R, int C) {
    __shared__ float tile[32][33];
    const int batch = blockIdx.z;
    const float* src = in + (size_t)batch * R * C;
    float* dst = out + (size_t)batch * R * C;
    const int c0 = blockIdx.x * 32;
    const int r0 = blockIdx.y * 32;
#pragma unroll
    for (int i = threadIdx.y; i < 32; i += 8)
        tile[i][threadIdx.x] = src[(size_t)(r0 + i) * C + (c0 + threadIdx.x)];
    __syncthreads();
#pragma unroll
    for (int i = threadIdx.y; i < 32; i += 8)
        dst[(size_t)(c0 + i) * R + (r0 + threadIdx.x)] = tile[threadIdx.x][i];
}

// ---------------------------------------------------------------------------
// Pre-pack offset_w [18][128][3][3] into per-K-step, per-lane WMMA B layout:
// step s = kk*32 + cb/4 owns 128 contiguous floats:
//   [0..63]  tile0 (N=0..15):  lane l gets B rows {koff, koff+1}, col l&15
//   [64..127] tile1 (N=16..31, zero-padded past 18)
// ---------------------------------------------------------------------------
__global__ void pack_b_kernel(const float* __restrict__ wo,
                              float* __restrict__ bp) {
    const int idx = blockIdx.x * 256 + threadIdx.x;  // 0..BPACK_N-1
    const int s    = idx >> 7;
    const int r    = idx & 127;
    const int tile = r >> 6;
    const int l    = (r >> 1) & 31;
    const int j    = r & 1;
    const int kk   = s >> 5;               // 0..8  (kh*3+kw)
    const int cb   = (s & 31) << 2;        // c-block base
    const int n    = (l & 15) + (tile << 4);
    const int c    = cb + ((l >> 4) << 1) + j;
    const float v  = (n < NOUT_) ? wo[(size_t)n * KTOT_ + (size_t)c * 9 + kk]
                                 : 0.f;
    bp[idx] = v;
}

// ---------------------------------------------------------------------------
// Offset conv as fp32 WMMA GEMM:  M=32768 pixels, K=1152, N=18 (pad 32).
// Packed B staged into LDS once per block via the Tensor Data Mover
// (tensor_load_to_lds + s_wait_tensorcnt), then the K-loop is:
//   global_load_b64 (A, im2col from NHWC) + 2x ds_load_b64 (B) + 2x WMMA.
// grid = 256 blocks * 8 waves = 2048 M-tiles; 147456 B dynamic LDS.
// ---------------------------------------------------------------------------
__global__ void offset_conv_wmma(const float* __restrict__ xn,     // [B][S][S][C]
                                 const float* __restrict__ bpack,  // [BPACK_N]
                                 const float* __restrict__ bias,   // [18]
                                 float* __restrict__ offs) {       // [NPIX][18]
    extern __shared__ float bsh[];  // BPACK_N floats

#if HAVE_TDM
    if (threadIdx.x < 32) {  // wave 0 issues one TDM DMA for the whole block
        const unsigned long long ga = (unsigned long long)(size_t)bpack;
        v4u g0;
        g0.x = 1u;                                   // count=1 (valid D#)
        g0.y = (unsigned)(size_t)bsh;                // lds_addr
        g0.z = (unsigned)(ga & 0xFFFFFFFFull);       // global_addr[31:0]
        g0.w = (unsigned)((ga >> 32) & 0x1FFFFFFull) // global_addr[56:32]
             | (2u << 30);                           // type=2 ("image")
        v8i g1;
        g1[0] = (int)0x00020000u;                          // data_size=4B
        g1[1] = (int)(((unsigned)BPACK_N & 0xFFFFu) << 16);// tensor_dim0 lo16
        g1[2] = (int)((((unsigned)BPACK_N >> 16) & 0xFFFFu)
                      | (1u << 16));                       // dim0 hi | dim1=1
        g1[3] = (int)(((unsigned)BPACK_N & 0xFFFFu) << 16);// tile_dim0=36864
        g1[4] = 1;                                         // tile_dim1=1
        g1[5] = (int)BPACK_N;                              // dim0_stride lo32
        g1[6] = 0;
        g1[7] = 0;
        v4i gz4 = {0, 0, 0, 0};
        v8i gz8 = {0, 0, 0, 0, 0, 0, 0, 0};
        __builtin_amdgcn_tensor_load_to_lds(g0, g1, gz4, gz4, gz8, 0);
        __builtin_amdgcn_s_wait_tensorcnt((short)0);
    }
#else
    for (int i = threadIdx.x; i < BPACK_N; i += 256) bsh[i] = bpack[i];
#endif
    __syncthreads();

    const int lane   = threadIdx.x & 31;
    const int tile   = blockIdx.x * 8 + (threadIdx.x >> 5);  // 0..2047
    const int row    = lane & 15;        // A row (M)
    const int hiHalf = lane >> 4;        // 0: K rows 0,1   1: K rows 2,3
    const int koff   = hiHalf * 2;

    const int p = tile * 16 + row;       // pixel for this lane's A row
    const int b = p >> 12;
    const int h = (p >> 6) & 63;
    const int w = p & 63;

    v8f acc0 = {};  // N = 0..15
    v8f acc1 = {};  // N = 16..31 (only 16,17 real)

    for (int kh = 0; kh < 3; ++kh) {
        const int y = h + kh - 1;
        for (int kw = 0; kw < 3; ++kw) {
            const int xw  = w + kw - 1;
            const bool inb = (unsigned)y < (unsigned)S_ &&
                             (unsigned)xw < (unsigned)S_;
            const float* asrc =
                xn + (((size_t)b * S_ + (inb ? y : 0)) * S_ + (inb ? xw : 0)) * C_;
            const int kk = kh * 3 + kw;
            const float* bstep = bsh + (size_t)kk * 32 * 128 + lane * 2;

            for (int cb4 = 0; cb4 < 32; ++cb4) {
                v2f a;
                if (inb) {
                    const v2f av = *(const v2f*)(asrc + cb4 * 4 + koff);
                    a = av;
                } else {
                    a.x = 0.f; a.y = 0.f;
                }
                const v2f b0 = *(const v2f*)(bstep + cb4 * 128);
                const v2f b1 = *(const v2f*)(bstep + cb4 * 128 + 64);
                acc0 = __builtin_amdgcn_wmma_f32_16x16x4_f32(
                    false, a, false, b0, (short)0, acc0, false, false);
                acc1 = __builtin_amdgcn_wmma_f32_16x16x4_f32(
                    false, a, false, b1, (short)0, acc1, false, false);
            }
        }
    }

    // D layout: VGPR v holds M = v + 8*hiHalf, N = lane&15 (+16 for acc1)
    const int n0 = row;
    const int n1 = row + 16;
    const bool n1ok = n1 < NOUT_;
    const float bias0 = bias[n0];
    const float bias1 = n1ok ? bias[n1] : 0.f;
#pragma unroll
    for (int v = 0; v < 8; ++v) {
        const int m  = v + hiHalf * 8;
        const int pp = tile * 16 + m;
        offs[(size_t)pp * NOUT_ + n0] = acc0[v] + bias0;
        if (n1ok) offs[(size_t)pp * NOUT_ + n1] = acc1[v] + bias1;
    }
}

// ---------------------------------------------------------------------------
// Deformable gather: one wave per pixel; lane owns 4 contiguous channels
// (NHWC -> float4 loads per corner, corner addresses wave-uniform).
// ---------------------------------------------------------------------------
__global__ void deform_gather(const float* __restrict__ xn,    // [B][S][S][C]
                              const float* __restrict__ offs,  // [NPIX][18]
                              const float* __restrict__ wk,    // [C][9]
                              float* __restrict__ outn) {      // [NPIX][C]
    const int lane = threadIdx.x & 31;
    const int p    = blockIdx.x * 8 + (threadIdx.x >> 5);
    const int b = p >> 12;
    const int h = (p >> 6) & 63;
    const int w = p & 63;
    const int c = lane * 4;

    float wkr[9][4];
#pragma unroll
    for (int t = 0; t < 9; ++t)
#pragma unroll
        for (int j = 0; j < 4; ++j)
            wkr[t][j] = wk[(size_t)(c + j) * 9 + t];

    const float* op = offs + (size_t)p * NOUT_;
    const float bx = -1.f + (2.f / 63.f) * (float)w;
    const float by = -1.f + (2.f / 63.f) * (float)h;
    const float* xb = xn + (size_t)b * HW_ * C_;

    float acc0 = 0.f, acc1 = 0.f, acc2 = 0.f, acc3 = 0.f;

#pragma unroll
    for (int t = 0; t < 9; ++t) {
        const float gx = ((op[t]     + bx + 1.f) * (float)S_ - 1.f) * 0.5f;
        const float gy = ((op[9 + t] + by + 1.f) * (float)S_ - 1.f) * 0.5f;
        const float x0f = floorf(gx), y0f = floorf(gy);
        const float wx1 = gx - x0f, wx0 = 1.f - wx1;
        const float wy1 = gy - y0f, wy0 = 1.f - wy1;
        const int ix0 = (int)x0f, iy0 = (int)y0f;
        const int ix1 = ix0 + 1,  iy1 = iy0 + 1;

        float s0 = 0.f, s1 = 0.f, s2 = 0.f, s3 = 0.f;
#define CORNER(IX, IY, WGT)                                                   \
        if ((unsigned)(IX) < (unsigned)S_ && (unsigned)(IY) < (unsigned)S_) { \
            const float4 v = *(const float4*)(xb +                            \
                (((size_t)(IY) * S_ + (size_t)(IX)) * C_ + c));               \
            s0 += (WGT) * v.x; s1 += (WGT) * v.y;                             \
            s2 += (WGT) * v.z; s3 += (WGT) * v.w;                             \
        }
        CORNER(ix0, iy0, wx0 * wy0)
        CORNER(ix1, iy0, wx1 * wy0)
        CORNER(ix0, iy1, wx0 * wy1)
        CORNER(ix1, iy1, wx1 * wy1)
#undef CORNER
        acc0 += wkr[t][0] * s0;
        acc1 += wkr[t][1] * s1;
        acc2 += wkr[t][2] * s2;
        acc3 += wkr[t][3] * s3;
    }

    float4 o;
    o.x = acc0; o.y = acc1; o.z = acc2; o.w = acc3;
    *(float4*)(outn + (size_t)p * C_ + c) = o;
}

// ---------------------------------------------------------------------------
extern "C" void kernel_launch(void* const* d_in, const int* in_sizes, int n_in,
                              void* d_out, int out_size, void* d_ws, size_t ws_size,
                              hipStream_t stream) {
    (void)in_sizes; (void)n_in; (void)out_size; (void)ws_size;

    const float* x        = (const float*)d_in[0];  // [8][128][64][64]
    const float* offset_w = (const float*)d_in[1];  // [18][128][3][3]
    const float* offset_b = (const float*)d_in[2];  // [18]
    const float* weight_w = (const float*)d_in[3];  // [1][128][3][3] == wk[128][9]
    float* out = (float*)d_out;                     // [8][128][64][64]

    // workspace layout (16B aligned chunks): 16MB + 2.25MB + 16MB + 144KB
    float* x_nhwc   = (float*)d_ws;                         // [8][64][64][128]
    float* offs     = x_nhwc + (size_t)NPIX_ * C_;          // [32768][18]
    float* out_nhwc = offs + (size_t)NPIX_ * NOUT_;         // [32768][128]
    float* bpack    = out_nhwc + (size_t)NPIX_ * C_;        // [BPACK_N]

    // 1) x: NCHW -> NHWC (per batch transpose of [128][4096])
    transpose_kernel<<<dim3(HW_ / 32, C_ / 32, B_), dim3(32, 8), 0, stream>>>(
        x, x_nhwc, C_, HW_);

    // 1b) pack B into per-lane WMMA layout
    pack_b_kernel<<<dim3(BPACK_N / 256), dim3(256), 0, stream>>>(
        offset_w, bpack);

    // 2) offset conv via fp32 WMMA GEMM; packed B staged to LDS via TDM
    offset_conv_wmma<<<dim3(256), dim3(256), BPACK_N * sizeof(float), stream>>>(
        x_nhwc, bpack, offset_b, offs);

    // 3) bilinear gather + tap-weighted accumulation (NHWC)
    deform_gather<<<dim3(NPIX_ / 8), dim3(256), 0, stream>>>(
        x_nhwc, offs, weight_w, out_nhwc);

    // 4) out: NHWC -> NCHW (per batch transpose of [4096][128])
    transpose_kernel<<<dim3(C_ / 32, HW_ / 32, B_), dim3(32, 8), 0, stream>>>(
        out_nhwc, out, HW_, C_);
}